// Discriminator_67027259621837
// MI455X (gfx1250) — compile-verified
//
#include <hip/hip_runtime.h>

typedef float v2f __attribute__((ext_vector_type(2)));
typedef float v8f __attribute__((ext_vector_type(8)));

#define WIDTH    40
#define LDSTRIDE 52   // row stride in floats: halves of the wave hit disjoint banks
#define KT       10   // K tiles of 4  (40 = 10*4, exact)
#define JT       3    // N tiles of 16 (40 padded to 48)
#define NPAIRS   20   // N_MID/2 (sin, softplus) pairs

__device__ __forceinline__ void middle_layer(float* __restrict__ tile,
                                             const v2f (&Bf)[KT][JT],
                                             const float (&bmv)[JT],
                                             int h, int lc, int act) {
    // A fragments: lane holds row lc, VGPR0/1 = k = 4t+2h / 4t+2h+1 -> one b64 load
    v2f A[KT];
#pragma unroll
    for (int t = 0; t < KT; ++t)
        A[t] = *(const v2f*)&tile[lc * LDSTRIDE + 4 * t + 2 * h];

    v8f D[JT];
#pragma unroll
    for (int j = 0; j < JT; ++j) {
        v8f c = {};
#pragma unroll
        for (int t = 0; t < KT; ++t)
            c = __builtin_amdgcn_wmma_f32_16x16x4_f32(
                    false, A[t], false, Bf[t][j], (short)0, c, false, false);
        D[j] = c;
    }

    // bias + activation + store back (wave-private tile: no barrier needed)
#pragma unroll
    for (int j = 0; j < JT; ++j) {
#pragma unroll
        for (int v = 0; v < 8; ++v) {
            float z = D[j][v] + bmv[j];
            if (act)  // softplus = log1p(exp(z)) with overflow guard
                z = (z > 20.f) ? z : __logf(1.f + __expf(z));
            else
                z = __sinf(z);
            tile[(v + 8 * h) * LDSTRIDE + 16 * j + lc] = z;
        }
    }
}

__global__ __launch_bounds__(256) void mlp40_kernel(
    const float* __restrict__ x,  const float* __restrict__ y,
    const float* __restrict__ W0, const float* __restrict__ b0,
    const float* __restrict__ Wm, const float* __restrict__ bm,
    const float* __restrict__ Wl, const float* __restrict__ bl,
    float* __restrict__ out, int n)
{
    __shared__ float smem[8 * 16 * LDSTRIDE];   // 8 waves * 3328 B = 26.6 KB
    const int lane = threadIdx.x & 31;
    const int wave = threadIdx.x >> 5;
    const int h    = lane >> 4;      // which 16-lane half
    const int lc   = lane & 15;
    float* tile = smem + wave * (16 * LDSTRIDE);

    // ---- preload resident weights (reused across many tiles) ----
    // B fragment (4x16 f32): VGPR p, lane l -> (k = p + 2h, n = lc)
    v2f Bf[KT][JT];
#pragma unroll
    for (int t = 0; t < KT; ++t)
#pragma unroll
        for (int j = 0; j < JT; ++j) {
            const int col = 16 * j + lc;
            const int k0  = 4 * t + 2 * h;
            float w0v = (col < WIDTH) ? Wm[(k0 + 0) * WIDTH + col] : 0.f;
            float w1v = (col < WIDTH) ? Wm[(k0 + 1) * WIDTH + col] : 0.f;
            Bf[t][j].x = w0v;
            Bf[t][j].y = w1v;
        }

    float bmv[JT], w00[JT], w01[JT], b0v[JT];
#pragma unroll
    for (int j = 0; j < JT; ++j) {
        const int col = 16 * j + lc;
        bmv[j] = (col < WIDTH) ? bm[col] : 0.f;
        w00[j] = (col < WIDTH) ? W0[0 * WIDTH + col] : 0.f;
        w01[j] = (col < WIDTH) ? W0[1 * WIDTH + col] : 0.f;
        b0v[j] = (col < WIDTH) ? b0[col] : 0.f;
    }
    float wl0[KT], wl1[KT];
#pragma unroll
    for (int i = 0; i < KT; ++i) {   // this lane reduces k = 20h .. 20h+19
        wl0[i] = Wl[20 * h + 2 * i + 0];
        wl1[i] = Wl[20 * h + 2 * i + 1];
    }
    const float blv = bl[0];

    const int numTiles = (n + 15) >> 4;
    const int gwave    = blockIdx.x * 8 + wave;
    const int nwaves   = gridDim.x * 8;

    for (int tidx = gwave; tidx < numTiles; tidx += nwaves) {
        const int base = tidx << 4;

        // ---- layer 0: [x,y] @ W0 + b0, tanh; write 16x48 tile (pad cols -> 0)
#pragma unroll
        for (int v = 0; v < 8; ++v) {
            const int m  = v + 8 * h;
            int gi = base + m; gi = (gi < n) ? gi : (n - 1);
            const float xv = x[gi], yv = y[gi];
#pragma unroll
            for (int j = 0; j < JT; ++j) {
                float z = fmaf(xv, w00[j], fmaf(yv, w01[j], b0v[j]));
                tile[m * LDSTRIDE + 16 * j + lc] = tanhf(z);
            }
        }

        // ---- 20 (sin, softplus) weight-shared pairs on the matrix pipe ----
#pragma unroll 1
        for (int p = 0; p < NPAIRS; ++p) {
            middle_layer(tile, Bf, bmv, h, lc, 0);  // sin
            middle_layer(tile, Bf, bmv, h, lc, 1);  // softplus
        }

        // ---- final 40 -> 1: split-K over the two lane-halves ----
        float acc = 0.f;
#pragma unroll
        for (int i = 0; i < KT; ++i) {
            v2f aa = *(const v2f*)&tile[lc * LDSTRIDE + 20 * h + 2 * i];
            acc = fmaf(aa.x, wl0[i], acc);
            acc = fmaf(aa.y, wl1[i], acc);
        }
        acc += __shfl_xor(acc, 16, 32);
        if (h == 0 && base + lc < n)
            out[base + lc] = acc + blv;
    }
}

extern "C" void kernel_launch(void* const* d_in, const int* in_sizes, int n_in,
                              void* d_out, int out_size, void* d_ws, size_t ws_size,
                              hipStream_t stream) {
    const float* x  = (const float*)d_in[0];
    const float* y  = (const float*)d_in[1];
    const float* W0 = (const float*)d_in[2];
    const float* b0 = (const float*)d_in[3];
    const float* Wm = (const float*)d_in[4];
    const float* bm = (const float*)d_in[5];
    const float* Wl = (const float*)d_in[6];
    const float* bl = (const float*)d_in[7];
    float* out = (float*)d_out;

    const int n = in_sizes[0];
    const int numTiles = (n + 15) / 16;
    int blocks = (numTiles + 7) / 8;          // 8 waves per block
    if (blocks > 2048) blocks = 2048;         // grid-stride over ~7.6 tiles/wave
    if (blocks < 1) blocks = 1;

    mlp40_kernel<<<blocks, 256, 0, stream>>>(x, y, W0, b0, Wm, bm, Wl, bl, out, n);
}